// Backbone_VideoMamba_13030930776655
// MI455X (gfx1250) — compile-verified
//
#include <hip/hip_runtime.h>
#include <hip/hip_bf16.h>

// ---------------------------------------------------------------------------
// VideoMamba backbone forward on gfx1250 (CDNA5), f32 end-to-end.
// GEMMs use V_WMMA_F32_16X16X4_F32 (wave32 matrix pipe, full f32 precision).
// A-slab (64 rows) is staged through LDS once per block per 32-wide K chunk
// via GLOBAL_LOAD_ASYNC_TO_LDS_B128 (ASYNCcnt); B chunk is register-
// prefetched; A fragments are software-pipelined one k-step ahead so the
// WMMA stream overlaps ds_load latency.
// ---------------------------------------------------------------------------

typedef __attribute__((ext_vector_type(2))) float v2f;
typedef __attribute__((ext_vector_type(4))) float v4f;
typedef __attribute__((ext_vector_type(8))) float v8f;
typedef __attribute__((ext_vector_type(4))) int   i4;

typedef i4 __attribute__((address_space(1)))* gptr_i4;   // global int4*
typedef i4 __attribute__((address_space(3)))* sptr_i4;   // LDS int4*

#if defined(__has_builtin)
#if __has_builtin(__builtin_amdgcn_global_load_async_to_lds_b128) && \
    __has_builtin(__builtin_amdgcn_s_wait_asynccnt)
#define USE_ASYNC_LDS 1
#endif
#endif
#ifndef USE_ASYNC_LDS
#define USE_ASYNC_LDS 0
#endif

// ---- model constants ----
#define BN    16      // batch
#define DM    384     // model dim
#define NDEPTH 24
#define IMGSZ 224
#define PATCH 16
#define NPATCH 196
#define LR    197     // real sequence length
#define LPAD  208     // padded to multiple of 16
#define DI    768     // inner dim
#define SN    16      // state dim
#define DTRK  24      // dt rank (real)
#define DTKP  32      // dt rank padded to K-chunk
#define DXP   56      // DTR + 2*S (real x-proj width)
#define XDP   64      // padded xdbl / xpw width
#define XZST  1536    // xz stride (2*DI)
#define PK    768     // patch GEMM K = 3*16*16

#define KCHUNK 32
#define LDSTR  36     // LDS row stride (floats): 16B-aligned rows, bank-conflict-free

// ---------------------------------------------------------------------------
// GEMM: Out[m, n] = sum_k A[m*lda + k] * W[n*ldw + k] + bias[n]
// Block = 128 threads (4 waves).  Block tile = 64 rows x 64 cols.
// Wave computes 4 M-tiles x 1 N-tile (64x16), A fragments from LDS.
// Requirements: K % 32 == 0, N % 64 == 0, M % 64 == 0.
// ---------------------------------------------------------------------------
__global__ __launch_bounds__(128)
void gemm_f32_wmma(const float* __restrict__ A, int lda,
                   const float* __restrict__ W, int ldw,
                   float* __restrict__ Out, int ldo,
                   int K, const float* __restrict__ bias)
{
    __shared__ float lAs[64 * LDSTR];

    const int tid  = threadIdx.x;
    const int lane = tid & 31;
    const int wave = tid >> 5;
    const int tn = (blockIdx.x * 4 + wave) * 16;
    const int tm = blockIdx.y * 64;
    const int half = lane >> 4;   // 0 or 1
    const int l15  = lane & 15;
    const int koff = half * 2;

    const float* Wr = W + (size_t)(tn + l15) * ldw + koff;
    const float* Ab = A + (size_t)tm * lda;

    // per-thread cooperative-fill coordinates (4 segments of 4 floats each)
    const int row0 = tid >> 3;              // 0..15 (+16 per step)
    const int seg0 = (tid & 7) * 4;         // 0,4,...,28

    // LDS fragment base pointers for the 4 M-tiles
    const float* l0 = lAs + (l15)      * LDSTR + koff;
    const float* l1 = lAs + (16 + l15) * LDSTR + koff;
    const float* l2 = lAs + (32 + l15) * LDSTR + koff;
    const float* l3 = lAs + (48 + l15) * LDSTR + koff;

    v8f acc0 = {}, acc1 = {}, acc2 = {}, acc3 = {};

    for (int kc = 0; kc < K; kc += KCHUNK) {
        // ---- prefetch this wave's B chunk into registers (global, b64) ----
        v2f breg[KCHUNK / 4];
        #pragma unroll
        for (int i = 0; i < KCHUNK / 4; ++i)
            breg[i] = *(const v2f*)(Wr + kc + i * 4);

        // ---- cooperative A-slab load: 64 rows x 32 cols -> LDS ----
        __syncthreads();   // previous chunk fully consumed
#if USE_ASYNC_LDS
        #pragma unroll
        for (int i = 0; i < 4; ++i) {
            const int row = row0 + i * 16;
            const float* gsrc = Ab + (size_t)row * lda + kc + seg0;
            __builtin_amdgcn_global_load_async_to_lds_b128(
                (gptr_i4)(__attribute__((address_space(1))) void*)(void*)const_cast<float*>(gsrc),
                (sptr_i4)(__attribute__((address_space(3))) void*)(void*)(lAs + row * LDSTR + seg0),
                0, 0);
        }
        __builtin_amdgcn_s_wait_asynccnt(0);
#else
        {
            v4f t0 = *(const v4f*)(Ab + (size_t)(row0)      * lda + kc + seg0);
            v4f t1 = *(const v4f*)(Ab + (size_t)(row0 + 16) * lda + kc + seg0);
            v4f t2 = *(const v4f*)(Ab + (size_t)(row0 + 32) * lda + kc + seg0);
            v4f t3 = *(const v4f*)(Ab + (size_t)(row0 + 48) * lda + kc + seg0);
            *(v4f*)(lAs + (row0)      * LDSTR + seg0) = t0;
            *(v4f*)(lAs + (row0 + 16) * LDSTR + seg0) = t1;
            *(v4f*)(lAs + (row0 + 32) * LDSTR + seg0) = t2;
            *(v4f*)(lAs + (row0 + 48) * LDSTR + seg0) = t3;
        }
#endif
        __syncthreads();

        // ---- 8 k-steps x 4 M-tiles of WMMA, A fragments pipelined 1 ahead ----
        v2f a0c = *(const v2f*)(l0);
        v2f a1c = *(const v2f*)(l1);
        v2f a2c = *(const v2f*)(l2);
        v2f a3c = *(const v2f*)(l3);
        #pragma unroll
        for (int i = 0; i < KCHUNK / 4; ++i) {
            v2f a0n = {}, a1n = {}, a2n = {}, a3n = {};
            if (i < KCHUNK / 4 - 1) {
                const int kk = (i + 1) * 4;
                a0n = *(const v2f*)(l0 + kk);
                a1n = *(const v2f*)(l1 + kk);
                a2n = *(const v2f*)(l2 + kk);
                a3n = *(const v2f*)(l3 + kk);
            }
            acc0 = __builtin_amdgcn_wmma_f32_16x16x4_f32(false, a0c, false, breg[i], (short)0, acc0, false, false);
            acc1 = __builtin_amdgcn_wmma_f32_16x16x4_f32(false, a1c, false, breg[i], (short)0, acc1, false, false);
            acc2 = __builtin_amdgcn_wmma_f32_16x16x4_f32(false, a2c, false, breg[i], (short)0, acc2, false, false);
            acc3 = __builtin_amdgcn_wmma_f32_16x16x4_f32(false, a3c, false, breg[i], (short)0, acc3, false, false);
            a0c = a0n; a1c = a1n; a2c = a2n; a3c = a3n;
        }
    }

    const int nr = tn + l15;
    const float bs = bias ? bias[nr] : 0.0f;
    float* o = Out + (size_t)(tm + half * 8) * ldo + nr;
    #pragma unroll
    for (int i = 0; i < 8; ++i) o[(size_t)i * ldo] = acc0[i] + bs;
    o += (size_t)16 * ldo;
    #pragma unroll
    for (int i = 0; i < 8; ++i) o[(size_t)i * ldo] = acc1[i] + bs;
    o += (size_t)16 * ldo;
    #pragma unroll
    for (int i = 0; i < 8; ++i) o[(size_t)i * ldo] = acc2[i] + bs;
    o += (size_t)16 * ldo;
    #pragma unroll
    for (int i = 0; i < 8; ++i) o[(size_t)i * ldo] = acc3[i] + bs;
}

// ---------------------------------------------------------------------------
// Pad x-proj weights (all layers): (24, 56, 768) -> (24, 64, 768), zero tail.
// ---------------------------------------------------------------------------
__global__ void pad_xpw_kernel(const float* __restrict__ src, float* __restrict__ dst)
{
    int idx = blockIdx.x * blockDim.x + threadIdx.x;
    const int total = NDEPTH * XDP * DI;
    if (idx >= total) return;
    int k  = idx % DI;
    int t  = idx / DI;
    int n  = t % XDP;
    int ly = t / XDP;
    dst[idx] = (n < DXP) ? src[((size_t)ly * DXP + n) * DI + k] : 0.0f;
}

// ---------------------------------------------------------------------------
// Pad dt-proj weights: (24*768, 24) -> (24*768, 32), zero tail columns.
// ---------------------------------------------------------------------------
__global__ void pad_dtw_kernel(const float* __restrict__ src, float* __restrict__ dst)
{
    int idx = blockIdx.x * blockDim.x + threadIdx.x;
    const int total = NDEPTH * DI * DTKP;
    if (idx >= total) return;
    int k = idx % DTKP;
    int r = idx / DTKP;
    dst[idx] = (k < DTRK) ? src[(size_t)r * DTRK + k] : 0.0f;
}

// ---------------------------------------------------------------------------
// im2col for patch embedding: patches[(b*196+p), c*256+py*16+px]
// ---------------------------------------------------------------------------
__global__ void im2col_kernel(const float* __restrict__ x, float* __restrict__ patches)
{
    int idx = blockIdx.x * blockDim.x + threadIdx.x;
    const int total = BN * NPATCH * PK;
    if (idx >= total) return;
    int k = idx % PK;
    int m = idx / PK;
    int p = m % NPATCH;
    int b = m / NPATCH;
    int c  = k / 256;
    int r  = k % 256;
    int py = r / 16;
    int px = r % 16;
    int ph = p / 14, pw = p % 14;
    int iy = ph * 16 + py, ix = pw * 16 + px;
    patches[(size_t)m * PK + k] =
        x[(((size_t)b * 3 + c) * IMGSZ + iy) * IMGSZ + ix];
}

// ---------------------------------------------------------------------------
// Assemble embedding into padded hid buffer: cls + patches(+bias already) + pos
// ---------------------------------------------------------------------------
__global__ void assemble_kernel(const float* __restrict__ pe,
                                const float* __restrict__ cls,
                                const float* __restrict__ pos,
                                float* __restrict__ hid)
{
    int idx = blockIdx.x * blockDim.x + threadIdx.x;
    const int total = BN * LR * DM;
    if (idx >= total) return;
    int dd = idx % DM;
    int t  = idx / DM;
    int li = t % LR;
    int b  = t / LR;
    float v;
    if (li == 0) v = cls[dd];
    else         v = pe[((size_t)b * NPATCH + (li - 1)) * DM + dd];
    hid[((size_t)b * LPAD + li) * DM + dd] = v + pos[(size_t)li * DM + dd];
}

// ---------------------------------------------------------------------------
// res += hid ; h = LayerNorm(res) * w + b      (one 384-thread block per token)
// ---------------------------------------------------------------------------
__global__ __launch_bounds__(DM)
void add_ln_kernel(const float* __restrict__ hid, float* __restrict__ res,
                   const float* __restrict__ w, const float* __restrict__ b,
                   float* __restrict__ h)
{
    int row = blockIdx.x;              // 0 .. B*LR-1
    int bi = row / LR, li = row % LR;
    size_t base = ((size_t)bi * LPAD + li) * DM;
    int t = threadIdx.x;
    int lane = t & 31, wv = t >> 5;

    float v = res[base + t] + hid[base + t];
    res[base + t] = v;

    __shared__ float s1[DM / 32];
    __shared__ float s2[DM / 32];

    float s = v;
    #pragma unroll
    for (int off = 16; off > 0; off >>= 1) s += __shfl_xor(s, off, 32);
    if (lane == 0) s1[wv] = s;
    __syncthreads();
    float tot = 0.f;
    #pragma unroll
    for (int i = 0; i < DM / 32; ++i) tot += s1[i];
    float mean = tot * (1.0f / DM);

    float dv = v - mean;
    float q = dv * dv;
    #pragma unroll
    for (int off = 16; off > 0; off >>= 1) q += __shfl_xor(q, off, 32);
    if (lane == 0) s2[wv] = q;
    __syncthreads();
    float tot2 = 0.f;
    #pragma unroll
    for (int i = 0; i < DM / 32; ++i) tot2 += s2[i];
    float var = tot2 * (1.0f / DM);

    h[base + t] = dv * rsqrtf(var + 1e-5f) * w[t] + b[t];
}

// ---------------------------------------------------------------------------
// Final LN: out = LN(res + hid) * fw + fb  -> unpadded (B, LR, DM)
// ---------------------------------------------------------------------------
__global__ __launch_bounds__(DM)
void final_ln_kernel(const float* __restrict__ hid, const float* __restrict__ res,
                     const float* __restrict__ w, const float* __restrict__ b,
                     float* __restrict__ out)
{
    int row = blockIdx.x;
    int bi = row / LR, li = row % LR;
    size_t base = ((size_t)bi * LPAD + li) * DM;
    int t = threadIdx.x;
    int lane = t & 31, wv = t >> 5;

    float v = res[base + t] + hid[base + t];

    __shared__ float s1[DM / 32];
    __shared__ float s2[DM / 32];

    float s = v;
    #pragma unroll
    for (int off = 16; off > 0; off >>= 1) s += __shfl_xor(s, off, 32);
    if (lane == 0) s1[wv] = s;
    __syncthreads();
    float tot = 0.f;
    #pragma unroll
    for (int i = 0; i < DM / 32; ++i) tot += s1[i];
    float mean = tot * (1.0f / DM);

    float dv = v - mean;
    float q = dv * dv;
    #pragma unroll
    for (int off = 16; off > 0; off >>= 1) q += __shfl_xor(q, off, 32);
    if (lane == 0) s2[wv] = q;
    __syncthreads();
    float tot2 = 0.f;
    #pragma unroll
    for (int i = 0; i < DM / 32; ++i) tot2 += s2[i];
    float var = tot2 * (1.0f / DM);

    out[((size_t)bi * LR + li) * DM + t] = dv * rsqrtf(var + 1e-5f) * w[t] + b[t];
}

// ---------------------------------------------------------------------------
// Causal depthwise conv (DCONV=4) + SiLU for forward and reversed sequences.
// x = xz[..., :DI].  Backward output stored in reversed order.
// ---------------------------------------------------------------------------
__global__ void conv_silu_kernel(const float* __restrict__ xz,
                                 const float* __restrict__ cw, const float* __restrict__ cb,
                                 const float* __restrict__ cwb, const float* __restrict__ cbb,
                                 float* __restrict__ xcf, float* __restrict__ xcbr)
{
    int idx = blockIdx.x * blockDim.x + threadIdx.x;
    const int total = BN * LR * DI;
    if (idx >= total) return;
    int d  = idx % DI;
    int t  = idx / DI;
    int li = t % LR;
    int bi = t / LR;

    float acc = cb[d];
    float accb = cbb[d];
    #pragma unroll
    for (int j = 0; j < 4; ++j) {
        int ls = li - 3 + j;
        float uf = 0.f, ub = 0.f;
        if (ls >= 0) {
            uf = xz[((size_t)bi * LPAD + ls) * XZST + d];
            ub = xz[((size_t)bi * LPAD + (LR - 1 - ls)) * XZST + d];
        }
        acc  += cw[d * 4 + j]  * uf;
        accb += cwb[d * 4 + j] * ub;
    }
    size_t o = ((size_t)bi * LPAD + li) * DI + d;
    xcf[o]  = acc  / (1.f + __expf(-acc));
    xcbr[o] = accb / (1.f + __expf(-accb));
}

// ---------------------------------------------------------------------------
// Selective scan + D-skip + SiLU(z) gate.  One wave32 handles 2 channels
// (16 states in each half-wave); sequential over LR timesteps.
// reverse=0: y written at li.  reverse=1: y accumulated at LR-1-li.
// ---------------------------------------------------------------------------
__global__ __launch_bounds__(256)
void scan_kernel(const float* __restrict__ xc, const float* __restrict__ dtraw,
                 const float* __restrict__ xdbl, const float* __restrict__ Alog,
                 const float* __restrict__ Dp, const float* __restrict__ xz,
                 float* __restrict__ ycomb, int reverse)
{
    int lane = threadIdx.x & 31;
    int wg = blockIdx.x * (blockDim.x >> 5) + (threadIdx.x >> 5);
    int sub = lane >> 4;
    int s   = lane & 15;
    int ch  = wg * 2 + sub;
    if (ch >= BN * DI) return;
    int bi = ch / DI, d = ch % DI;

    const float Av = -__expf(Alog[(size_t)d * SN + s]);
    const float Dv = Dp[d];
    float h = 0.f;

    for (int li = 0; li < LR; ++li) {
        size_t r = (size_t)bi * LPAD + li;
        float u    = xc[r * DI + d];
        float dtv  = dtraw[r * DI + d];
        dtv = (dtv > 20.f) ? dtv : log1pf(__expf(dtv));   // softplus
        float Bv = xdbl[r * XDP + DTRK + s];
        float Cv = xdbl[r * XDP + DTRK + SN + s];
        h = h * __expf(dtv * Av) + (dtv * u) * Bv;
        float p = h * Cv;
        p += __shfl_xor(p, 1, 32);
        p += __shfl_xor(p, 2, 32);
        p += __shfl_xor(p, 4, 32);
        p += __shfl_xor(p, 8, 32);
        if (s == 0) {
            int lo = reverse ? (LR - 1 - li) : li;
            float y = p + u * Dv;
            float zv = xz[((size_t)bi * LPAD + lo) * XZST + DI + d];
            y *= zv / (1.f + __expf(-zv));
            size_t o = ((size_t)bi * LPAD + lo) * DI + d;
            if (reverse) ycomb[o] += y;
            else         ycomb[o]  = y;
        }
    }
}

// ---------------------------------------------------------------------------
// Host side
// ---------------------------------------------------------------------------
extern "C" void kernel_launch(void* const* d_in, const int* in_sizes, int n_in,
                              void* d_out, int out_size, void* d_ws, size_t ws_size,
                              hipStream_t stream)
{
    const float* x        = (const float*)d_in[0];
    const float* patch_w  = (const float*)d_in[1];
    const float* patch_b  = (const float*)d_in[2];
    const float* cls_tok  = (const float*)d_in[3];
    const float* pos_emb  = (const float*)d_in[4];
    const float* norm_w   = (const float*)d_in[5];
    const float* norm_b   = (const float*)d_in[6];
    const float* in_w     = (const float*)d_in[7];
    const float* cw       = (const float*)d_in[8];
    const float* cb       = (const float*)d_in[9];
    const float* xpw      = (const float*)d_in[10];
    const float* dtw      = (const float*)d_in[11];
    const float* dtb      = (const float*)d_in[12];
    const float* A_log    = (const float*)d_in[13];
    const float* Dp       = (const float*)d_in[14];
    const float* cwb      = (const float*)d_in[15];
    const float* cbb      = (const float*)d_in[16];
    const float* xpwb     = (const float*)d_in[17];
    const float* dtwb     = (const float*)d_in[18];
    const float* dtbb     = (const float*)d_in[19];
    const float* A_logb   = (const float*)d_in[20];
    const float* Dpb      = (const float*)d_in[21];
    const float* out_w    = (const float*)d_in[22];
    const float* fw       = (const float*)d_in[23];
    const float* fb       = (const float*)d_in[24];
    float* out = (float*)d_out;

    // ---- workspace carving (floats) ----
    float* ws = (float*)d_ws;
    const size_t M  = (size_t)BN * LPAD;           // 3328 padded rows
    float* xz       = ws;                          // M*1536
    float* xc_f     = xz       + M * XZST;         // M*768
    float* xc_b     = xc_f     + M * DI;           // M*768
    float* dt_f     = xc_b     + M * DI;           // M*768
    float* dt_b     = dt_f     + M * DI;           // M*768
    float* ycomb    = dt_b     + M * DI;           // M*768
    float* xdbl_f   = ycomb    + M * DI;           // M*64
    float* xdbl_b   = xdbl_f   + M * XDP;          // M*64
    float* hid      = xdbl_b   + M * XDP;          // M*384
    float* res      = hid      + M * DM;           // M*384
    float* hbuf     = res      + M * DM;           // M*384
    float* xpw_pad  = hbuf     + M * DM;           // 24*64*768
    float* xpwb_pad = xpw_pad  + (size_t)NDEPTH * XDP * DI;
    float* dtw_pad  = xpwb_pad + (size_t)NDEPTH * XDP * DI;  // 24*768*32
    float* dtwb_pad = dtw_pad  + (size_t)NDEPTH * DI * DTKP;
    // stage-1 buffers aliased onto later-stage scratch:
    float* patches = xz;                           // (B*196)*768 <= M*1536
    float* pe      = xc_f;                         // (B*196)*384 <= M*768

    auto gemm = [&](const float* A, int lda, const float* W, int ldw,
                    float* O, int ldo, int Mrows, int N, int K, const float* bias) {
        dim3 grid(N / 64, Mrows / 64);
        hipLaunchKernelGGL(gemm_f32_wmma, grid, dim3(128), 0, stream,
                           A, lda, W, ldw, O, ldo, K, bias);
    };

    // ---- pre-pad x-proj and dt-proj weights for all layers ----
    {
        int tot = NDEPTH * XDP * DI;
        hipLaunchKernelGGL(pad_xpw_kernel, dim3((tot + 255) / 256), dim3(256), 0, stream,
                           xpw, xpw_pad);
        hipLaunchKernelGGL(pad_xpw_kernel, dim3((tot + 255) / 256), dim3(256), 0, stream,
                           xpwb, xpwb_pad);
        int tot2 = NDEPTH * DI * DTKP;
        hipLaunchKernelGGL(pad_dtw_kernel, dim3((tot2 + 255) / 256), dim3(256), 0, stream,
                           dtw, dtw_pad);
        hipLaunchKernelGGL(pad_dtw_kernel, dim3((tot2 + 255) / 256), dim3(256), 0, stream,
                           dtwb, dtwb_pad);
    }

    // ---- patch embedding ----
    {
        int tot = BN * NPATCH * PK;
        hipLaunchKernelGGL(im2col_kernel, dim3((tot + 255) / 256), dim3(256), 0, stream,
                           x, patches);
        gemm(patches, PK, patch_w, PK, pe, DM, BN * NPATCH, DM, PK, patch_b);
        int tot2 = BN * LR * DM;
        hipLaunchKernelGGL(assemble_kernel, dim3((tot2 + 255) / 256), dim3(256), 0, stream,
                           pe, cls_tok, pos_emb, hid);
        (void)hipMemsetAsync(res, 0, M * DM * sizeof(float), stream);
    }

    // ---- 24 Mamba blocks ----
    for (int ly = 0; ly < NDEPTH; ++ly) {
        const float* nw_  = norm_w   + (size_t)ly * DM;
        const float* nb_  = norm_b   + (size_t)ly * DM;
        const float* inw_ = in_w     + (size_t)ly * XZST * DM;
        const float* cw_  = cw       + (size_t)ly * DI * 4;
        const float* cb_  = cb       + (size_t)ly * DI;
        const float* xpwp_  = xpw_pad  + (size_t)ly * XDP * DI;
        const float* xpwbp_ = xpwb_pad + (size_t)ly * XDP * DI;
        const float* dtwp_  = dtw_pad  + (size_t)ly * DI * DTKP;
        const float* dtwbp_ = dtwb_pad + (size_t)ly * DI * DTKP;
        const float* dtb_ = dtb      + (size_t)ly * DI;
        const float* Al_  = A_log    + (size_t)ly * DI * SN;
        const float* Dp_  = Dp       + (size_t)ly * DI;
        const float* cwb_ = cwb      + (size_t)ly * DI * 4;
        const float* cbb_ = cbb      + (size_t)ly * DI;
        const float* dtbb_= dtbb     + (size_t)ly * DI;
        const float* Alb_ = A_logb   + (size_t)ly * DI * SN;
        const float* Dpb_ = Dpb      + (size_t)ly * DI;
        const float* ow_  = out_w    + (size_t)ly * DM * DI;

        // res += hid; h = LN(res)
        hipLaunchKernelGGL(add_ln_kernel, dim3(BN * LR), dim3(DM), 0, stream,
                           hid, res, nw_, nb_, hbuf);

        // xz = h @ in_w.T   (M x 1536, K=384)
        gemm(hbuf, DM, inw_, DM, xz, XZST, (int)M, XZST, DM, nullptr);

        // depthwise causal conv + SiLU, fwd + reversed
        {
            int tot = BN * LR * DI;
            hipLaunchKernelGGL(conv_silu_kernel, dim3((tot + 255) / 256), dim3(256), 0, stream,
                               xz, cw_, cb_, cwb_, cbb_, xc_f, xc_b);
        }

        // xdbl = xc @ xpw_pad.T  (M x 64, K=768), cols 56..63 are zero
        gemm(xc_f, DI, xpwp_,  DI, xdbl_f, XDP, (int)M, XDP, DI, nullptr);
        gemm(xc_b, DI, xpwbp_, DI, xdbl_b, XDP, (int)M, XDP, DI, nullptr);

        // dt_raw = xdbl[:, :32] @ dtw_pad.T + dtb  (M x 768, K=32; cols 24..31
        // of the padded weight are zero so the extra xdbl columns contribute 0)
        gemm(xdbl_f, XDP, dtwp_,  DTKP, dt_f, DI, (int)M, DI, DTKP, dtb_);
        gemm(xdbl_b, XDP, dtwbp_, DTKP, dt_b, DI, (int)M, DI, DTKP, dtbb_);

        // selective scans (+ gate); backward accumulates into un-reversed slots
        {
            dim3 grid((BN * DI / 2 + 7) / 8);  // 8 waves/block, 2 ch/wave
            hipLaunchKernelGGL(scan_kernel, grid, dim3(256), 0, stream,
                               xc_f, dt_f, xdbl_f, Al_, Dp_, xz, ycomb, 0);
            hipLaunchKernelGGL(scan_kernel, grid, dim3(256), 0, stream,
                               xc_b, dt_b, xdbl_b, Alb_, Dpb_, xz, ycomb, 1);
        }

        // hid = ycomb @ out_w.T  (M x 384, K=768)
        gemm(ycomb, DI, ow_, DI, hid, DM, (int)M, DM, DI, nullptr);
    }

    // ---- final LN -> d_out (unpadded) ----
    hipLaunchKernelGGL(final_ln_kernel, dim3(BN * LR), dim3(DM), 0, stream,
                       hid, res, fw, fb, out);
}